// MultiResolutionHashGrid_3702261809601
// MI455X (gfx1250) — compile-verified
//
#include <hip/hip_runtime.h>
#include <math.h>

// MultiResolutionHashGrid forward (Instant-NGP style) for MI455X / gfx1250.
//
// Thread layout: tid = pt*16 + lvl  (16 levels per point, wave32-friendly:
// each wave covers 2 points; 16 consecutive lanes emit 16 consecutive
// float2 outputs -> 256B contiguous NT store per wave).
//
// CDNA5-specific paths used:
//  - global_load_async_to_lds_b128 + s_wait_asynccnt (ASYNCcnt) to stage the
//    block's 16 positions (192B = 12 x 16B chunks) into LDS once, instead of
//    16x redundant per-lane loads.
//  - __builtin_nontemporal_store on the 128MB output stream (th:TH_STORE_NT)
//    so the write-once output does not evict the 64MB hash table from the
//    192MB L2 (the gathers are L2-resident, which is the whole game here).

typedef float v2f __attribute__((ext_vector_type(2)));

struct Res16 { float r[16]; };

#define TBITS 19u
#define TMASK ((1u << TBITS) - 1u)
#define PRIME1 2654435761u
#define PRIME2 805459861u

__global__ __launch_bounds__(256) void hashgrid_fwd_kernel(
    const float* __restrict__ pos,   // [n,3] f32
    const float* __restrict__ tab,   // [16, 2^19, 2] f32
    float* __restrict__ out,         // [n, 32] f32
    int n, Res16 R)
{
    __shared__ float sp[48];     // 16 points x 3 floats
    __shared__ float sres[16];   // per-level resolutions

    const int pb = blockIdx.x * 16;          // first point of this block
    const bool full = (pb + 16) <= n;        // full block of 16 points?

    if (threadIdx.x == 0) {
#pragma unroll
        for (int i = 0; i < 16; ++i) sres[i] = R.r[i];  // constant indices -> s_loads
    }

    if (full) {
        if (threadIdx.x < 12) {
            // 192 bytes = 12 lanes x 16B; block base is 16B-aligned (192 = 12*16).
            unsigned long long ga = (unsigned long long)(size_t)pos
                                  + (unsigned long long)pb * 12ull
                                  + (unsigned long long)threadIdx.x * 16ull;
            unsigned int ldsoff = (unsigned int)(size_t)(&sp[0])
                                + threadIdx.x * 16u;
            asm volatile("global_load_async_to_lds_b128 %0, %1, off"
                         :: "v"(ldsoff), "v"(ga) : "memory");
        }
        asm volatile("s_wait_asynccnt 0" ::: "memory");
    }
    __syncthreads();

    const long long t = (long long)blockIdx.x * 256 + threadIdx.x;
    if (t >= (long long)n * 16) return;
    const int pt  = (int)(t >> 4);
    const int lvl = threadIdx.x & 15;

    float px, py, pz;
    if (full) {
        const int lp = (threadIdx.x >> 4) * 3;
        px = sp[lp + 0]; py = sp[lp + 1]; pz = sp[lp + 2];
    } else {
        const float* p = pos + (size_t)pt * 3;
        px = p[0]; py = p[1]; pz = p[2];
    }

    // normalized = clip((p+1)/2.00000001, 0, 1-1e-6); the f32 divisor rounds
    // to exactly 2.0f, so (p+1)*0.5f is bit-identical.
    const float CLIP = (float)(1.0 - 1e-6);
    float nx = fminf(fmaxf((px + 1.0f) * 0.5f, 0.0f), CLIP);
    float ny = fminf(fmaxf((py + 1.0f) * 0.5f, 0.0f), CLIP);
    float nz = fminf(fmaxf((pz + 1.0f) * 0.5f, 0.0f), CLIP);

    const float res = sres[lvl];
    float sx = nx * res, sy = ny * res, sz = nz * res;
    float fxx = floorf(sx), fyy = floorf(sy), fzz = floorf(sz);
    float fx = sx - fxx, fy = sy - fyy, fz = sz - fzz;
    unsigned int x0 = (unsigned int)(int)fxx;
    unsigned int y0 = (unsigned int)(int)fyy;
    unsigned int z0 = (unsigned int)(int)fzz;

    // Hash components: (x+ox)*1 ^ (y+oy)*P1 ^ (z+oz)*P2, mod 2^19.
    unsigned int hx0 = x0,           hx1 = x0 + 1u;
    unsigned int hy0 = y0 * PRIME1,  hy1 = hy0 + PRIME1;
    unsigned int hz0 = z0 * PRIME2,  hz1 = hz0 + PRIME2;

    const unsigned int base = ((unsigned int)lvl) << TBITS;
    const v2f* __restrict__ T2 = (const v2f*)tab;

    // Corner order matches OFFSETS: index bits (x<<2)|(y<<1)|z
    unsigned int i000 = base + ((hx0 ^ hy0 ^ hz0) & TMASK);
    unsigned int i001 = base + ((hx0 ^ hy0 ^ hz1) & TMASK);
    unsigned int i010 = base + ((hx0 ^ hy1 ^ hz0) & TMASK);
    unsigned int i011 = base + ((hx0 ^ hy1 ^ hz1) & TMASK);
    unsigned int i100 = base + ((hx1 ^ hy0 ^ hz0) & TMASK);
    unsigned int i101 = base + ((hx1 ^ hy0 ^ hz1) & TMASK);
    unsigned int i110 = base + ((hx1 ^ hy1 ^ hz0) & TMASK);
    unsigned int i111 = base + ((hx1 ^ hy1 ^ hz1) & TMASK);

    // 8 independent 8-byte gathers (global_load_b64), L2-resident table.
    v2f f000 = T2[i000];
    v2f f001 = T2[i001];
    v2f f010 = T2[i010];
    v2f f011 = T2[i011];
    v2f f100 = T2[i100];
    v2f f101 = T2[i101];
    v2f f110 = T2[i110];
    v2f f111 = T2[i111];

    float wz0 = 1.0f - fz;
    v2f c00 = f000 * wz0 + f001 * fz;
    v2f c01 = f010 * wz0 + f011 * fz;
    v2f c10 = f100 * wz0 + f101 * fz;
    v2f c11 = f110 * wz0 + f111 * fz;
    float wy0 = 1.0f - fy;
    v2f c0 = c00 * wy0 + c01 * fy;
    v2f c1 = c10 * wy0 + c11 * fy;
    float wx0 = 1.0f - fx;
    v2f o = c0 * wx0 + c1 * fx;

    // Coalesced streaming store: lanes of a wave cover 32 consecutive float2s.
    __builtin_nontemporal_store(o, (v2f*)out + ((size_t)pt * 16 + lvl));
}

extern "C" void kernel_launch(void* const* d_in, const int* in_sizes, int n_in,
                              void* d_out, int out_size, void* d_ws, size_t ws_size,
                              hipStream_t stream)
{
    const float* pos = (const float*)d_in[0];   // positions [N,3] f32
    const float* tab = (const float*)d_in[1];   // hash_tables [16, 2^19, 2] f32
    float* out = (float*)d_out;                 // [N, 32] f32
    const int n = in_sizes[0] / 3;

    // Replicate Python's resolution table bit-for-bit with the same libm
    // double ops (exp/log/pow/ceil). Level 15 sits at 2048 +/- ~1e-11, so it
    // must be computed, not hardcoded.
    Res16 R;
    const double growth = exp((log(2048.0) - log(16.0)) / 15.0);
    for (int l = 0; l < 16; ++l)
        R.r[l] = (float)(int)ceil(16.0 * pow(growth, (double)l));

    const long long threads = (long long)n * 16;
    const int grid = (int)((threads + 255) / 256);
    hashgrid_fwd_kernel<<<grid, 256, 0, stream>>>(pos, tab, out, n, R);
    (void)d_ws; (void)ws_size; (void)n_in; (void)out_size;
}